// Event_Encoder_44083544326599
// MI455X (gfx1250) — compile-verified
//
#include <hip/hip_runtime.h>
#include <hip/hip_bf16.h>

#define N_NODES 100000
#define N_EDGES 1250000
#define HGCN    64
#define HIDDEN  256
#define ALPHA   0.1f

typedef __attribute__((ext_vector_type(2))) float v2f;
typedef __attribute__((ext_vector_type(8))) float v8f;

// ---------------------------------------------------------------- utilities
__global__ void zero_kernel(float* __restrict__ p, int n) {
    int i = blockIdx.x * blockDim.x + threadIdx.x;
    if (i < n) p[i] = 0.0f;
}

__global__ void deg_kernel(const long long* __restrict__ src, float* __restrict__ deg) {
    int e = blockIdx.x * blockDim.x + threadIdx.x;
    if (e < N_EDGES) unsafeAtomicAdd(&deg[(int)src[e]], 1.0f);
}

__global__ void dinv_kernel(const float* __restrict__ deg, float* __restrict__ dinv) {
    int i = blockIdx.x * blockDim.x + threadIdx.x;
    if (i < N_NODES) {
        float d = deg[i];
        dinv[i] = (d > 0.0f) ? rsqrtf(d) : 0.0f;
    }
}

__global__ void norm_kernel(const long long* __restrict__ ei,
                            const float* __restrict__ dinv,
                            float* __restrict__ nrm) {
    int e = blockIdx.x * blockDim.x + threadIdx.x;
    if (e < N_EDGES) nrm[e] = dinv[(int)ei[e]] * dinv[(int)ei[N_EDGES + e]];
}

// ----------------------------------------------------- fused MLP with WMMA
// One wave owns a 16-row strip of x. Both GEMMs run on the f32 WMMA pipe
// (V_WMMA_F32_16X16X4_F32, exact fp32). Each loop iteration feeds TWO
// independent accumulator chains (two 16-col output tiles sharing the A
// operand) so consecutive v_wmma ops have no RAW dependency and the XDL
// pipeline stays at issue rate instead of latency rate.
//
// f32 WMMA operand layout (wave32):
//   A 16x4 : lane = (K>=2)*16 + M, vgpr v = K&1     -> a = x[row=lane%16][k+2*(lane>>4)+v]
//   B 4x16 : lane = (K>=2)*16 + N, vgpr v selects K -> b = W[k+2*(lane>>4)+v][col=lane%16]
//   C/D    : vgpr g, lanes 0-15 -> row g, lanes 16-31 -> row g+8; col = lane%16
//
// GEMM1 result (ReLU'd 16x256 strip) is staged in LDS with stride 257 so the
// column-direction A reads of GEMM2 are bank-conflict free. GEMM2 epilogue
// writes both h and the final-output seed out = ALPHA*h (saves a full pass).
#define MLP_WAVES 2
__global__ __launch_bounds__(MLP_WAVES * 32) void mlp_wmma_kernel(
        const float* __restrict__ x,  const float* __restrict__ W1,
        const float* __restrict__ b1, const float* __restrict__ W2,
        const float* __restrict__ b2, float* __restrict__ h,
        float* __restrict__ out) {
    __shared__ float lds_h1[MLP_WAVES][16][HIDDEN + 1];

    const int wave = threadIdx.x >> 5;
    const int lane = threadIdx.x & 31;
    const int lm   = lane & 15;   // row (A) / col (B,D) within tile
    const int lk   = lane >> 4;   // selects K pair {0,1} vs {2,3}
    const int row_base = (blockIdx.x * MLP_WAVES + wave) * 16;

    // ---- preload this lane's full A strip for GEMM1 (16 K-steps x v2f)
    const float* xrow = x + (long)(row_base + lm) * HGCN;
    v2f areg[HGCN / 4];
#pragma unroll
    for (int kk = 0; kk < HGCN / 4; ++kk)
        areg[kk] = *(const v2f*)(xrow + kk * 4 + 2 * lk);

    // ---- GEMM1: h1 = relu(x @ W1 + b1) -> LDS ; two N-tiles per pass
    for (int t = 0; t < HIDDEN / 16; t += 2) {
        const int nb0 = t * 16;
        const int nb1 = nb0 + 16;
        v8f c0 = {};
        v8f c1 = {};
#pragma unroll
        for (int kk = 0; kk < HGCN / 4; ++kk) {
            const int ka = kk * 4 + 2 * lk;
            const v2f a = areg[kk];
            v2f b0, b1v;
            b0.x  = W1[(long)(ka + 0) * HIDDEN + nb0 + lm];
            b0.y  = W1[(long)(ka + 1) * HIDDEN + nb0 + lm];
            b1v.x = W1[(long)(ka + 0) * HIDDEN + nb1 + lm];
            b1v.y = W1[(long)(ka + 1) * HIDDEN + nb1 + lm];
            c0 = __builtin_amdgcn_wmma_f32_16x16x4_f32(false, a, false, b0,
                                                       (short)0, c0, false, false);
            c1 = __builtin_amdgcn_wmma_f32_16x16x4_f32(false, a, false, b1v,
                                                       (short)0, c1, false, false);
        }
        const float bias0 = b1[nb0 + lm];
        const float bias1 = b1[nb1 + lm];
#pragma unroll
        for (int g = 0; g < 8; ++g) {
            float v0 = c0[g] + bias0;
            float v1 = c1[g] + bias1;
            lds_h1[wave][g + 8 * lk][nb0 + lm] = (v0 > 0.0f) ? v0 : 0.0f;
            lds_h1[wave][g + 8 * lk][nb1 + lm] = (v1 > 0.0f) ? v1 : 0.0f;
        }
    }
    __syncthreads();

    // ---- GEMM2: h = h1 @ W2 + b2 -> global ; two N-tiles per pass
    for (int t = 0; t < HGCN / 16; t += 2) {
        const int nb0 = t * 16;
        const int nb1 = nb0 + 16;
        v8f c0 = {};
        v8f c1 = {};
#pragma unroll 8
        for (int k = 0; k < HIDDEN; k += 4) {
            const int ka = k + 2 * lk;
            v2f a;
            a.x = lds_h1[wave][lm][ka + 0];
            a.y = lds_h1[wave][lm][ka + 1];
            v2f b0, b1v;
            b0.x  = W2[(long)(ka + 0) * HGCN + nb0 + lm];
            b0.y  = W2[(long)(ka + 1) * HGCN + nb0 + lm];
            b1v.x = W2[(long)(ka + 0) * HGCN + nb1 + lm];
            b1v.y = W2[(long)(ka + 1) * HGCN + nb1 + lm];
            c0 = __builtin_amdgcn_wmma_f32_16x16x4_f32(false, a, false, b0,
                                                       (short)0, c0, false, false);
            c1 = __builtin_amdgcn_wmma_f32_16x16x4_f32(false, a, false, b1v,
                                                       (short)0, c1, false, false);
        }
        const float bias0 = b2[nb0 + lm];
        const float bias1 = b2[nb1 + lm];
#pragma unroll
        for (int g = 0; g < 8; ++g) {
            const long row = row_base + g + 8 * lk;
            float v0 = c0[g] + bias0;
            float v1 = c1[g] + bias1;
            h[row * HGCN + nb0 + lm]   = v0;
            h[row * HGCN + nb1 + lm]   = v1;
            out[row * HGCN + nb0 + lm] = ALPHA * v0;   // seed: out = alpha*h
            out[row * HGCN + nb1 + lm] = ALPHA * v1;
        }
    }
}

// ------------------------------------------------------------- propagation
// 16 lanes per edge, float4 per lane (64 features). Gather is a contiguous
// b128 load from the L2-resident feature matrix; scatter uses hardware f32
// atomics (working set ~100 MB fits the 192 MB L2).
__global__ void prop_kernel(const long long* __restrict__ ei,
                            const float* __restrict__ nrm,
                            const float* __restrict__ vin,
                            float* __restrict__ vout,
                            float scale) {
    int idx = blockIdx.x * blockDim.x + threadIdx.x;  // E * 16 threads exactly
    int e   = idx >> 4;
    int c4  = (idx & 15) << 2;
    long long s = ei[e];
    long long d = ei[N_EDGES + e];
    float w = nrm[e] * scale;
    const float4 v = *(const float4*)(vin + s * HGCN + c4);
    float* o = vout + d * HGCN + c4;
    unsafeAtomicAdd(o + 0, w * v.x);
    unsafeAtomicAdd(o + 1, w * v.y);
    unsafeAtomicAdd(o + 2, w * v.z);
    unsafeAtomicAdd(o + 3, w * v.w);
}

// ------------------------------------------------------------------ launch
extern "C" void kernel_launch(void* const* d_in, const int* in_sizes, int n_in,
                              void* d_out, int out_size, void* d_ws, size_t ws_size,
                              hipStream_t stream) {
    const float*     x  = (const float*)d_in[0];
    const long long* ei = (const long long*)d_in[1];  // int64 [2, E]
    const float*     W1 = (const float*)d_in[2];
    const float*     b1 = (const float*)d_in[3];
    const float*     W2 = (const float*)d_in[4];
    const float*     b2 = (const float*)d_in[5];
    float*           out = (float*)d_out;

    float* ws   = (float*)d_ws;
    float* deg  = ws;                               // N
    float* dinv = deg  + N_NODES;                   // N
    float* nrm  = dinv + N_NODES;                   // E
    float* h    = nrm  + N_EDGES;                   // N*64 (16B aligned)
    float* p1   = h    + (size_t)N_NODES * HGCN;    // N*64 (16B aligned)

    const float g1 = ALPHA * (1.0f - ALPHA) * (1.0f - ALPHA);

    // deterministic per-call init of accumulators
    zero_kernel<<<(N_NODES + 255) / 256, 256, 0, stream>>>(deg, N_NODES);
    zero_kernel<<<(N_NODES * HGCN + 255) / 256, 256, 0, stream>>>(p1, N_NODES * HGCN);

    // degree -> dinv -> per-edge norm
    deg_kernel<<<(N_EDGES + 255) / 256, 256, 0, stream>>>(ei, deg);
    dinv_kernel<<<(N_NODES + 255) / 256, 256, 0, stream>>>(deg, dinv);
    norm_kernel<<<(N_EDGES + 255) / 256, 256, 0, stream>>>(ei, dinv, nrm);

    // fused MLP on the f32 WMMA pipe; epilogue also seeds out = alpha*h
    mlp_wmma_kernel<<<N_NODES / (16 * MLP_WAVES), MLP_WAVES * 32, 0, stream>>>(
        x, W1, b1, W2, b2, h, out);

    // p1 = A h, then out += g1 * A p1
    prop_kernel<<<(N_EDGES * 16) / 256, 256, 0, stream>>>(ei, nrm, h, p1, 1.0f);
    prop_kernel<<<(N_EDGES * 16) / 256, 256, 0, stream>>>(ei, nrm, p1, out, g1);
}